// GraphSim_26362509263243
// MI455X (gfx1250) — compile-verified
//
#include <hip/hip_runtime.h>
#include <hip/hip_bf16.h>

// Problem constants from the reference: R=8 rotations, N=1024 rows, F=256 feats.
#define R 8
#define N 1024
#define F 256
#define NPAIR (R * R)
#define EPS 1e-12f

typedef float v2f __attribute__((ext_vector_type(2)));
typedef float v8f __attribute__((ext_vector_type(8)));

// ---------------------------------------------------------------------------
// Kernel 1: per-row inverse L2 norm. One 256-thread block per row.
// ---------------------------------------------------------------------------
__global__ void __launch_bounds__(256) row_invnorm_kernel(
    const float* __restrict__ feat, float* __restrict__ inv) {
  const int row = blockIdx.x;            // 0 .. R*N-1
  const int tid = threadIdx.x;           // 0 .. 255 (== F)
  float x = feat[(size_t)row * F + tid];
  float ss = x * x;
#pragma unroll
  for (int off = 16; off > 0; off >>= 1) ss += __shfl_down(ss, off, 32);
  __shared__ float wsum[8];
  if ((tid & 31) == 0) wsum[tid >> 5] = ss;
  __syncthreads();
  if (tid == 0) {
    float s = 0.0f;
#pragma unroll
    for (int w = 0; w < 8; ++w) s += wsum[w];
    inv[row] = 1.0f / fmaxf(sqrtf(s), EPS);
  }
}

// ---------------------------------------------------------------------------
// Kernel 2: batched 8x256 @ 256x8 per-row diag matmul via V_WMMA_F32_16X16X4_F32.
// One wave handles one 16x16 tile = two packed rows (2t, 2t+1):
//   A fragment rows  M=0..7  -> rotations of row 2t,  M=8..15 -> row 2t+1
//   B fragment cols  N=0..7  -> rotations of row 2t,  N=8..15 -> row 2t+1
// Only the two diagonal 8x8 blocks of C are kept (cross-row blocks discarded).
// ISA 32-bit A 16x4 layout: lanes 0-15 hold (K=k0,k0+1), lanes 16-31 (k0+2,k0+3)
// -> one aligned float2 load per operand per K-step. B mirrors A with N<->M.
// ---------------------------------------------------------------------------
__global__ void __launch_bounds__(256) pairdot_wmma_kernel(
    const float* __restrict__ A, const float* __restrict__ B,
    const float* __restrict__ inva, const float* __restrict__ invb,
    float* __restrict__ V) {
  const int lane = threadIdx.x & 31;
  const int wave = threadIdx.x >> 5;
  const int tile = blockIdx.x * 8 + wave;  // 0 .. 511
  const int r0 = tile * 2;
  const int r1 = r0 + 1;

  const int m = lane & 15;                 // M for A / N for B
  const int khalf = (lane >> 4) << 1;      // 0 or 2
  const int rot = m & 7;
  const int row = (m < 8) ? r0 : r1;

  const float* pa = A + ((size_t)(rot * N + row)) * F + khalf;
  const float* pb = B + ((size_t)(rot * N + row)) * F + khalf;

  v8f c = {0.f, 0.f, 0.f, 0.f, 0.f, 0.f, 0.f, 0.f};
#pragma unroll 8
  for (int k = 0; k < F; k += 4) {
    v2f a = *(const v2f*)pa;
    v2f b = *(const v2f*)pb;
    c = __builtin_amdgcn_wmma_f32_16x16x4_f32(
        /*neg_a=*/false, a, /*neg_b=*/false, b,
        /*c_mod=*/(short)0, c, /*reuse_a=*/false, /*reuse_b=*/false);
    pa += 4;
    pb += 4;
  }

  // C/D layout: element (M,Ncol) lives in VGPR (M&7), lane Ncol + 16*(M>>3).
  // Row r0 block: lanes 0-7 (j=lane), accumulator slot i.
  // Row r1 block: lanes 24-31 (j=lane-24), accumulator slot i.
  int j = -1, nrow = 0;
  if (lane < 8) {
    j = lane;
    nrow = r0;
  } else if (lane >= 24) {
    j = lane - 24;
    nrow = r1;
  }
  if (j >= 0) {
#pragma unroll
    for (int i = 0; i < 8; ++i) {
      float val = c[i] * inva[i * N + nrow] * invb[j * N + nrow];
      V[(size_t)(i * 8 + j) * N + nrow] = val;
    }
  }
}

// ---------------------------------------------------------------------------
// Kernel 3: one 512-thread block per (i,j) pair.
// Bitonic-sort v (descending) in LDS, then compute:
//   sim_feat = mean(sorted[0:16]);  w~_p = sorted_p - sorted[N-1]
//   e_p = ca_i[p] - cb_j[p]  (ORIGINAL index p, per reference semantics)
//   S = 2*(Sum w~)*(Sum w~_p |e_p|^2) - 2*|Sum w~_p e_p|^2
//   dis = S / (N^2 * max(Sum w~^2, eps));  out = sim_feat - dis
// ---------------------------------------------------------------------------
__global__ void __launch_bounds__(512) pair_reduce_kernel(
    const float* __restrict__ V, const float* __restrict__ ca,
    const float* __restrict__ cb, float* __restrict__ out) {
  const int pair = blockIdx.x;  // i*8 + j
  const int pi = pair >> 3;
  const int pj = pair & 7;
  const int t = threadIdx.x;  // 0 .. 511
  __shared__ float sv[N];
  __shared__ float red[512];

  sv[t] = V[(size_t)pair * N + t];
  sv[t + 512] = V[(size_t)pair * N + t + 512];
  __syncthreads();

  // Bitonic sort, descending. Each thread handles indices t and t+512;
  // compare pairs within a stage are disjoint, barrier between stages.
  for (int k = 2; k <= N; k <<= 1) {
    for (int jj = k >> 1; jj > 0; jj >>= 1) {
#pragma unroll
      for (int base = 0; base < N; base += 512) {
        int idx = base + t;
        int ixj = idx ^ jj;
        if (ixj > idx) {
          bool desc = ((idx & k) == 0);
          float x = sv[idx], y = sv[ixj];
          bool sw = desc ? (x < y) : (x > y);
          if (sw) {
            sv[idx] = y;
            sv[ixj] = x;
          }
        }
      }
      __syncthreads();
    }
  }

  const float vmin = sv[N - 1];
  float lW1 = 0.f, lW2 = 0.f, lQ = 0.f, lMX = 0.f, lMY = 0.f;
#pragma unroll
  for (int base = 0; base < N; base += 512) {
    int p = base + t;
    float w = sv[p] - vmin;  // sorted-position weight
    float ex = ca[(pi * N + p) * 2 + 0] - cb[(pj * N + p) * 2 + 0];
    float ey = ca[(pi * N + p) * 2 + 1] - cb[(pj * N + p) * 2 + 1];
    lW1 += w;
    lW2 += w * w;
    lQ += w * (ex * ex + ey * ey);
    lMX += w * ex;
    lMY += w * ey;
  }

  float vals[5] = {lW1, lW2, lQ, lMX, lMY};
  float res[5];
#pragma unroll
  for (int v = 0; v < 5; ++v) {
    red[t] = vals[v];
    __syncthreads();
    for (int off = 256; off > 0; off >>= 1) {
      if (t < off) red[t] += red[t + off];
      __syncthreads();
    }
    res[v] = red[0];
    __syncthreads();
  }

  if (t == 0) {
    float top = 0.f;
#pragma unroll
    for (int p = 0; p < 16; ++p) top += sv[p];
    float sim_feat = top * (1.0f / 16.0f);
    float W1 = res[0], W2 = res[1], Q = res[2], mx = res[3], my = res[4];
    float S = 2.0f * (W1 * Q - (mx * mx + my * my));
    float dis = S / ((float)N * (float)N * fmaxf(W2, EPS));
    out[pair] = sim_feat - dis;
  }
}

// ---------------------------------------------------------------------------
// Kernel 4: max over 64 pair scores -> scalar output.
// ---------------------------------------------------------------------------
__global__ void __launch_bounds__(64) max_kernel(const float* __restrict__ in,
                                                 float* __restrict__ out) {
  const int t = threadIdx.x;  // 0..63
  float v = in[t];
#pragma unroll
  for (int off = 16; off > 0; off >>= 1) v = fmaxf(v, __shfl_down(v, off, 32));
  __shared__ float w[2];
  if ((t & 31) == 0) w[t >> 5] = v;
  __syncthreads();
  if (t == 0) out[0] = fmaxf(w[0], w[1]);
}

extern "C" void kernel_launch(void* const* d_in, const int* in_sizes, int n_in,
                              void* d_out, int out_size, void* d_ws,
                              size_t ws_size, hipStream_t stream) {
  const float* A = (const float*)d_in[0];   // graph_feat_a (8,1024,256)
  const float* CA = (const float*)d_in[1];  // coord_a (8,1024,2)
  const float* B = (const float*)d_in[2];   // graph_feat_b (8,1024,256)
  const float* CB = (const float*)d_in[3];  // coord_b (8,1024,2)

  float* ws = (float*)d_ws;
  float* inva = ws;                       // R*N floats
  float* invb = inva + R * N;             // R*N floats
  float* V = invb + R * N;                // NPAIR*N floats
  float* pres = V + (size_t)NPAIR * N;    // NPAIR floats

  row_invnorm_kernel<<<R * N, 256, 0, stream>>>(A, inva);
  row_invnorm_kernel<<<R * N, 256, 0, stream>>>(B, invb);
  pairdot_wmma_kernel<<<64, 256, 0, stream>>>(A, B, inva, invb, V);
  pair_reduce_kernel<<<NPAIR, 512, 0, stream>>>(V, CA, CB, pres);
  max_kernel<<<1, 64, 0, stream>>>(pres, (float*)d_out);
}